// SNN_original_32186484916881
// MI455X (gfx1250) — compile-verified
//
#include <hip/hip_runtime.h>

// ---------------------------------------------------------------------------
// 3-layer LIF SNN forward on MI455X (gfx1250).
//
// Roofline: ~3.5 GFLOP vs ~370 MB streamed (x 168MB + mask 168MB + out 32MB)
// -> HBM floor ~16us @ 23.3 TB/s. Pure-VALU fp32 would be issue-bound at
// ~2-3x that, so the GEMMs run on the matrix pipe with V_WMMA_F32_16X16X4_F32
// (full fp32 semantics; I=H=20 is exactly 5 K-steps of 4). Only the ~220
// elementwise LIF ops per (b,t) stay on VALU.
//
//  * One wave owns a 16-row batch tile for all T steps; mem1/mem2/mem3 live
//    in WMMA C-layout accumulator registers across the whole run (B=8192 ->
//    512 waves).
//  * Leak term 0.5*mem is folded into the WMMA C seed. Layer 2 seeds C=0 so
//    the dropout mask can scale cur2 before the membrane update.
//  * Spike chaining (C-layout -> A-layout) via a per-wave LDS transpose
//    buffer, stride 36 floats (conflict-free for both access patterns).
//    LDS is in-order within a wave, so only compiler fences are needed.
//  * x loaded from global directly in A-layout (5 x b64/lane/t), mask in
//    C-layout; both double-buffered across t (prefetch t+1 during t) since
//    occupancy is ~1 wave/SIMD.
// ---------------------------------------------------------------------------

typedef __attribute__((ext_vector_type(2))) float v2f;
typedef __attribute__((ext_vector_type(8))) float v8f;

#define WAVES_PER_BLOCK 8
#define LDS_STRIDE 36  // floats per row; >32 cols, bank-conflict free

struct Buf {
  v2f ax[5];     // x tile in WMMA A-layout (K-steps 0..4)
  int mk[2][8];  // dropout mask in WMMA C-layout (2 N-tiles x 8 row-regs)
};

static __device__ __forceinline__ v8f wmma4(v2f a, v2f b, v8f c) {
  // D(16x16 f32) = A(16x4 f32) * B(4x16 f32) + C
  return __builtin_amdgcn_wmma_f32_16x16x4_f32(false, a, false, b, (short)0, c,
                                               false, false);
}

__global__ __launch_bounds__(WAVES_PER_BLOCK * 32)
void snn_lif_wmma(const float* __restrict__ x, const float* __restrict__ W1,
                  const float* __restrict__ W2, const float* __restrict__ W3,
                  const float* __restrict__ th1p, const float* __restrict__ th2p,
                  const float* __restrict__ th3p, const int* __restrict__ mask,
                  int T, int B, float* __restrict__ out_spk,
                  float* __restrict__ out_mem) {
  __shared__ float sbuf[WAVES_PER_BLOCK][16 * LDS_STRIDE];

  const int wave = threadIdx.x >> 5;
  const int lane = threadIdx.x & 31;
  const int l    = lane & 15;   // N (C/B layouts) or M (A layout)
  const int half = lane >> 4;   // half-wave select
  const int batchBase = (blockIdx.x * WAVES_PER_BLOCK + wave) * 16;
  if (batchBase >= B) return;  // no block-level syncs used -> safe early exit
  float* sw = sbuf[wave];

  const float th1 = th1p[0];
  const float th2 = th2p[0];
  const float th3 = th3p[0];

  // --- Weights staged once into WMMA B-layout registers. ------------------
  // B tile (4x16) for K-step k: VGPR pair holds rows {4k+2*half, 4k+2*half+1},
  // column N = l. Bmat[i][n] = W[n][i] (computing act @ W.T). N-pad -> 0.
  v2f bw1[2][5], bw2[2][5], bw3[5];
#pragma unroll
  for (int nt = 0; nt < 2; ++nt) {
    const int h  = nt * 16 + l;
    const bool v = (h < 20);
#pragma unroll
    for (int k = 0; k < 5; ++k) {
      const int i0 = 4 * k + 2 * half;
      bw1[nt][k].x = v ? W1[h * 20 + i0]     : 0.f;
      bw1[nt][k].y = v ? W1[h * 20 + i0 + 1] : 0.f;
      bw2[nt][k].x = v ? W2[h * 20 + i0]     : 0.f;
      bw2[nt][k].y = v ? W2[h * 20 + i0 + 1] : 0.f;
    }
  }
#pragma unroll
  for (int k = 0; k < 5; ++k) {
    const int i0 = 4 * k + 2 * half;
    bw3[k].x = (l < 2) ? W3[l * 20 + i0]     : 0.f;
    bw3[k].y = (l < 2) ? W3[l * 20 + i0 + 1] : 0.f;
  }

  // Membrane state in C-layout regs: reg r = batch row (r + 8*half), col l.
  float m1[2][8], m2[2][8], m3[8];
#pragma unroll
  for (int nt = 0; nt < 2; ++nt)
#pragma unroll
    for (int r = 0; r < 8; ++r) { m1[nt][r] = 0.f; m2[nt][r] = 0.f; }
#pragma unroll
  for (int r = 0; r < 8; ++r) m3[r] = 0.f;

  auto prefetch = [&](Buf& b, int t) {
    // x in A-layout: lane -> batch row l; half-wave selects K pair.
    const float* xrow = x + ((size_t)t * B + (batchBase + l)) * 20 + 2 * half;
#pragma unroll
    for (int k = 0; k < 5; ++k) b.ax[k] = *(const v2f*)(xrow + 4 * k);
    // mask in C-layout (per reg r: batch r+8*half, col n; clamp pad cols).
#pragma unroll
    for (int nt = 0; nt < 2; ++nt) {
      const int n = (nt == 0) ? l : ((l < 4) ? (16 + l) : 19);
      const int* mrow = mask + ((size_t)t * B + (batchBase + 8 * half)) * 20 + n;
#pragma unroll
      for (int r = 0; r < 8; ++r) b.mk[nt][r] = mrow[r * 20];
    }
  };

  auto step = [&](int t, const Buf& cur, Buf& nxt, bool pf) {
    if (pf) prefetch(nxt, t + 1);

    // ---- Layer 1: mem1 = 0.5*mem1 + x@W1.T - reset*th1 ------------------
#pragma unroll
    for (int nt = 0; nt < 2; ++nt) {
      v8f acc;
#pragma unroll
      for (int r = 0; r < 8; ++r) acc[r] = 0.5f * m1[nt][r];  // leak as C seed
#pragma unroll
      for (int k = 0; k < 5; ++k) acc = wmma4(cur.ax[k], bw1[nt][k], acc);
#pragma unroll
      for (int r = 0; r < 8; ++r) {
        const float rst = (m1[nt][r] > th1) ? th1 : 0.f;  // reset on OLD mem
        const float mn  = acc[r] - rst;
        m1[nt][r] = mn;
        const float spk = (mn > th1) ? 1.f : 0.f;
        if (nt == 0 || l < 4)
          sw[(r + 8 * half) * LDS_STRIDE + nt * 16 + l] = spk;
      }
    }
    __builtin_amdgcn_wave_barrier();

    // ---- Layer 2: cur2 = (spk1@W2.T) * mask*2 ---------------------------
    v2f a2[5];
#pragma unroll
    for (int k = 0; k < 5; ++k)
      a2[k] = *(const v2f*)&sw[l * LDS_STRIDE + 4 * k + 2 * half];
    v8f acc2[2];
#pragma unroll
    for (int nt = 0; nt < 2; ++nt) {
      v8f acc;
#pragma unroll
      for (int r = 0; r < 8; ++r) acc[r] = 0.f;  // C=0: mask scales cur2 only
#pragma unroll
      for (int k = 0; k < 5; ++k) acc = wmma4(a2[k], bw2[nt][k], acc);
      acc2[nt] = acc;
    }
    __builtin_amdgcn_wave_barrier();  // all spk1 reads before spk2 overwrite
#pragma unroll
    for (int nt = 0; nt < 2; ++nt) {
#pragma unroll
      for (int r = 0; r < 8; ++r) {
        const float mf  = cur.mk[nt][r] ? 2.f : 0.f;
        const float rst = (m2[nt][r] > th2) ? th2 : 0.f;
        const float mn  = 0.5f * m2[nt][r] + acc2[nt][r] * mf - rst;
        m2[nt][r] = mn;
        const float spk = (mn > th2) ? 1.f : 0.f;
        if (nt == 0 || l < 4)
          sw[(r + 8 * half) * LDS_STRIDE + nt * 16 + l] = spk;
      }
    }
    __builtin_amdgcn_wave_barrier();

    // ---- Layer 3: mem3 = 0.5*mem3 + spk2@W3.T - reset*th3 ---------------
    v2f a3[5];
#pragma unroll
    for (int k = 0; k < 5; ++k)
      a3[k] = *(const v2f*)&sw[l * LDS_STRIDE + 4 * k + 2 * half];
    v8f acc3;
#pragma unroll
    for (int r = 0; r < 8; ++r) acc3[r] = 0.5f * m3[r];
#pragma unroll
    for (int k = 0; k < 5; ++k) acc3 = wmma4(a3[k], bw3[k], acc3);
#pragma unroll
    for (int r = 0; r < 8; ++r) {
      const float rst = (m3[r] > th3) ? th3 : 0.f;
      const float mn  = acc3[r] - rst;
      m3[r] = mn;
      if (l < 2) {  // only O=2 valid output columns
        const size_t row = (size_t)t * B + (batchBase + r + 8 * half);
        out_spk[row * 2 + l] = (mn > th3) ? 1.f : 0.f;
        out_mem[row * 2 + l] = mn;
      }
    }
  };

  // Double-buffered time loop (2x unrolled so buffers stay in registers).
  Buf b0, b1;
  prefetch(b0, 0);
  int t = 0;
  for (; t + 2 <= T; t += 2) {
    step(t, b0, b1, true);
    step(t + 1, b1, b0, t + 2 < T);
  }
  if (t < T) step(t, b0, b1, false);
}

extern "C" void kernel_launch(void* const* d_in, const int* in_sizes, int n_in,
                              void* d_out, int out_size, void* d_ws,
                              size_t ws_size, hipStream_t stream) {
  (void)n_in; (void)out_size; (void)d_ws; (void)ws_size;
  const float* x    = (const float*)d_in[0];
  const float* W1   = (const float*)d_in[1];
  const float* W2   = (const float*)d_in[2];
  const float* W3   = (const float*)d_in[3];
  const float* th1  = (const float*)d_in[4];
  const float* th2  = (const float*)d_in[5];
  const float* th3  = (const float*)d_in[6];
  const int*   mask = (const int*)d_in[7];

  const int T = 256;                    // num_steps (setup_inputs)
  const int I = 20;
  const int B = in_sizes[0] / (T * I);  // 8192

  float* out_spk = (float*)d_out;
  float* out_mem = out_spk + (size_t)T * B * 2;

  const int waves  = (B + 15) / 16;
  const int blocks = (waves + WAVES_PER_BLOCK - 1) / WAVES_PER_BLOCK;
  snn_lif_wmma<<<blocks, WAVES_PER_BLOCK * 32, 0, stream>>>(
      x, W1, W2, W3, th1, th2, th3, mask, T, B, out_spk, out_mem);
}